// CenterLoss_64948495450317
// MI455X (gfx1250) — compile-verified
//
#include <hip/hip_runtime.h>

// CenterLoss on MI455X (gfx1250, wave32).
//
// picked[b] = ||x_b - centers[labels_b]||^2 ; out = mean(picked).
// Bandwidth-bound (~40MB @ 23.3TB/s); compute routed through
// V_WMMA_F32_16X16X4_F32: acc += diff * diff^T, diagonal = squared norms.
//
// f32 16x4 A-matrix VGPR layout (ISA 7.12.2):
//   lanes 0-15 : M = lane,    VGPR0 = K0, VGPR1 = K1
//   lanes 16-31: M = lane-16, VGPR0 = K2, VGPR1 = K3
// The 4x16 B-layout of diff^T places the *same* values in the same lanes,
// so one 2-VGPR pair serves as both A and B operands.

typedef __attribute__((ext_vector_type(2))) float v2f;
typedef __attribute__((ext_vector_type(8))) float v8f;

#define B_ROWS 16384
#define D_DIM  512
#define ROWS_PER_BLOCK 128   // 8 waves x 16 rows
#define THREADS 256
#define NBLOCKS (B_ROWS / ROWS_PER_BLOCK)  // 128

__global__ __launch_bounds__(THREADS)
void center_loss_partial(const float* __restrict__ x,
                         const int* __restrict__ labels,
                         const float* __restrict__ centers,
                         float* __restrict__ block_sums)
{
    __shared__ float wave_sums[THREADS / 32];

    const int lane = threadIdx.x & 31;
    const int wave = threadIdx.x >> 5;
    const int m    = lane & 15;            // row within 16-row tile
    const int koff = (lane >> 4) << 1;     // K sub-offset: 0 (lanes 0-15) or 2 (lanes 16-31)
    const int row  = blockIdx.x * ROWS_PER_BLOCK + wave * 16 + m;

    const int lab = labels[row];
    const float* xr = x       + (size_t)row * D_DIM + koff;
    const float* cr = centers + (size_t)lab * D_DIM + koff;

    v8f acc = {};
    #pragma unroll 8
    for (int k = 0; k < D_DIM; k += 4) {
        v2f xa = *(const v2f*)(xr + k);    // 8B aligned global_load_b64
        v2f ca = *(const v2f*)(cr + k);
        v2f a  = xa - ca;                  // diff chunk (2 of the 4 K values)
        // D = A * B + C with A = B = diff tile (16x4 f32)
        acc = __builtin_amdgcn_wmma_f32_16x16x4_f32(
            /*neg_a=*/false, a, /*neg_b=*/false, a,
            /*c_mod=*/(short)0, acc, /*reuse_a=*/false, /*reuse_b=*/false);
    }

    // Diagonal extraction: (M,M) lives at VGPR M, lane M (M<8)
    // and VGPR M-8, lane M+16 (M>=8). Lanes 8..23 contribute 0.
    const int idx = (lane < 8) ? lane : (lane - 24);  // negative => no match
    float v = 0.0f;
    #pragma unroll
    for (int i = 0; i < 8; ++i) {
        v = (idx == i) ? acc[i] : v;
    }

    // Deterministic wave32 butterfly reduction.
    #pragma unroll
    for (int off = 16; off > 0; off >>= 1) {
        v += __shfl_xor(v, off, 32);
    }
    if (lane == 0) wave_sums[wave] = v;
    __syncthreads();

    if (threadIdx.x == 0) {
        float s = 0.0f;
        #pragma unroll
        for (int w = 0; w < THREADS / 32; ++w) s += wave_sums[w];
        block_sums[blockIdx.x] = s;
    }
}

__global__ void center_loss_finalize(const float* __restrict__ block_sums,
                                     float* __restrict__ out, int nblocks)
{
    if (threadIdx.x == 0 && blockIdx.x == 0) {
        float s = 0.0f;
        for (int i = 0; i < nblocks; ++i) s += block_sums[i];  // fixed order
        out[0] = s * (1.0f / (float)B_ROWS);
    }
}

extern "C" void kernel_launch(void* const* d_in, const int* in_sizes, int n_in,
                              void* d_out, int out_size, void* d_ws, size_t ws_size,
                              hipStream_t stream) {
    const float* x       = (const float*)d_in[0];
    const int*   labels  = (const int*)d_in[1];   // harness: integer -> const int*
    const float* centers = (const float*)d_in[2];
    float* out        = (float*)d_out;
    float* block_sums = (float*)d_ws;             // 128 floats of scratch

    center_loss_partial<<<NBLOCKS, THREADS, 0, stream>>>(x, labels, centers, block_sums);
    center_loss_finalize<<<1, 32, 0, stream>>>(block_sums, out, NBLOCKS);
}